// GAT_8177617732164
// MI455X (gfx1250) — compile-verified
//
#include <hip/hip_runtime.h>
#include <math.h>

typedef __attribute__((ext_vector_type(2))) float v2f;
typedef __attribute__((ext_vector_type(8))) float v8f;

#define N_NODES 10000
#define E_EDGES 320000
#define ETOT    (E_EDGES + N_NODES)   // 330000, self-loops appended
#define IN_C    256
#define HIDC    128
#define HEADS   4
#define H1DIM   (HEADS * HIDC)        // 512
#define OUT_C   64
#define NGR     64
#define NEG_SLOPE 0.2f

#define CDIV(a,b) (((a)+(b)-1)/(b))

// ---------- helpers ----------
__device__ __forceinline__ unsigned ordf(float f) {
  unsigned u = __float_as_uint(f);
  return (u & 0x80000000u) ? ~u : (u | 0x80000000u);   // monotone map f -> u32
}
__device__ __forceinline__ float unordf(unsigned u) {
  return __uint_as_float((u & 0x80000000u) ? (u & 0x7fffffffu) : ~u);
}
__device__ __forceinline__ void edge_pair(const int* __restrict__ ei, int e, int& s, int& d) {
  if (e < E_EDGES) { s = ei[e]; d = ei[E_EDGES + e]; }
  else             { s = d = e - E_EDGES; }            // self loop
}

// ---------- fills (re-init every launch; harness does not re-poison) ----------
__global__ void fill_f32(float* __restrict__ p, float v, int n) {
  int t = blockIdx.x * blockDim.x + threadIdx.x;
  if (t < n) p[t] = v;
}
__global__ void fill_u32(unsigned* __restrict__ p, unsigned v, int n) {
  int t = blockIdx.x * blockDim.x + threadIdx.x;
  if (t < n) p[t] = v;
}

// ---------- WMMA f32 GEMM: C[M,NCOL] = A[M,K] @ B[K,NCOL] ----------
// One wave computes a 16x32 output strip (two 16x16 WMMA tiles sharing the A
// fragment -> two independent accumulator chains). Requires M % 16 == 0,
// NCOL % 32 == 0 (true for all GEMMs in this network: M = 10000 = 625*16).
template <int K, int NCOL>
__global__ __launch_bounds__(256)
void gemm_wmma_f32(const float* __restrict__ A, const float* __restrict__ B,
                   float* __restrict__ C, int M) {
  constexpr int COLT = NCOL / 32;
  int wave = (int)((blockIdx.x * blockDim.x + threadIdx.x) >> 5);
  int lane = threadIdx.x & 31;
  int rowTiles = M >> 4;
  if (wave >= rowTiles * COLT) return;                 // whole-wave exit, EXEC stays full
  int rowT = wave / COLT, colT = wave % COLT;
  int row0 = rowT << 4, col0 = colT << 5;

  int r   = lane & 15;
  int khi = (lane >> 4) << 1;                          // lanes 0-15: K{0,1}; 16-31: K{2,3}
  const float* Ap = A + (size_t)(row0 + r) * K + khi;  // 8B-aligned (K, khi even)
  const float* Bp = B + (size_t)khi * NCOL + col0 + r;

  v8f c0 = {}, c1 = {};
#pragma unroll 8
  for (int k0 = 0; k0 < K; k0 += 4) {
    v2f a = *(const v2f*)(Ap + k0);                    // global_load_b64
    v2f b0, b1;
    b0.x = Bp[(size_t)(k0    ) * NCOL];
    b0.y = Bp[(size_t)(k0 + 1) * NCOL];
    b1.x = Bp[(size_t)(k0    ) * NCOL + 16];
    b1.y = Bp[(size_t)(k0 + 1) * NCOL + 16];
    c0 = __builtin_amdgcn_wmma_f32_16x16x4_f32(false, a, false, b0, (short)0, c0, false, false);
    c1 = __builtin_amdgcn_wmma_f32_16x16x4_f32(false, a, false, b1, (short)0, c1, false, false);
  }
  // C/D layout: VGPR v -> M = v (+8 for upper lane half), N = lane&15
  int n  = lane & 15;
  int mb = (lane >> 4) << 3;
  float* Cp = C + (size_t)(row0 + mb) * NCOL + col0 + n;
#pragma unroll
  for (int v = 0; v < 8; ++v) {
    Cp[(size_t)v * NCOL]      = c0[v];
    Cp[(size_t)v * NCOL + 16] = c1[v];
  }
}

// ---------- per-(node,head) attention logits: one wave each ----------
__global__ __launch_bounds__(256)
void alpha_kernel(const float* __restrict__ h, const float* __restrict__ as,
                  const float* __restrict__ ad, float* __restrict__ outS,
                  float* __restrict__ outD, int nNodes, int heads, int C) {
  int wave = (int)((blockIdx.x * blockDim.x + threadIdx.x) >> 5);
  int lane = threadIdx.x & 31;
  if (wave >= nNodes * heads) return;
  int node = wave / heads, hd = wave % heads;
  const float* hp = h + (size_t)node * heads * C + (size_t)hd * C;
  float ss = 0.f, sd = 0.f;
  for (int i = lane; i < C; i += 32) {
    float v = hp[i];
    ss += v * as[hd * C + i];
    sd += v * ad[hd * C + i];
  }
  for (int off = 16; off; off >>= 1) {
    ss += __shfl_down(ss, off, 32);
    sd += __shfl_down(sd, off, 32);
  }
  if (lane == 0) { outS[wave] = ss; outD[wave] = sd; }
}

// ---------- edge softmax pass 1: segment max over destination ----------
__global__ void edge_max(const int* __restrict__ ei, const float* __restrict__ as,
                         const float* __restrict__ ad, unsigned* __restrict__ mx, int heads) {
  int t = blockIdx.x * blockDim.x + threadIdx.x;
  if (t >= ETOT * heads) return;
  int e = t / heads, hd = t % heads;
  int s, d; edge_pair(ei, e, s, d);
  float v = as[s * heads + hd] + ad[d * heads + hd];
  v = (v > 0.f) ? v : v * NEG_SLOPE;
  atomicMax(mx + d * heads + hd, ordf(v));
}

// ---------- pass 2: exp(e - max), accumulate denominator ----------
__global__ void edge_exp(const int* __restrict__ ei, const float* __restrict__ as,
                         const float* __restrict__ ad, const unsigned* __restrict__ mx,
                         float* __restrict__ w, float* __restrict__ den, int heads) {
  int t = blockIdx.x * blockDim.x + threadIdx.x;
  if (t >= ETOT * heads) return;
  int e = t / heads, hd = t % heads;
  int s, d; edge_pair(ei, e, s, d);
  float v = as[s * heads + hd] + ad[d * heads + hd];
  v = (v > 0.f) ? v : v * NEG_SLOPE;
  float ww = __expf(v - unordf(mx[d * heads + hd]));
  w[t] = ww;
  atomicAdd(den + d * heads + hd, ww);
}

// ---------- pass 3: weighted scatter-add, one wave per edge ----------
__global__ __launch_bounds__(256)
void edge_aggr(const int* __restrict__ ei, const float* __restrict__ h,
               const float* __restrict__ w, const float* __restrict__ den,
               float* __restrict__ acc, int heads, int C) {
  int wave = (int)((blockIdx.x * blockDim.x + threadIdx.x) >> 5);
  int lane = threadIdx.x & 31;
  if (wave >= ETOT) return;
  int s, d; edge_pair(ei, wave, s, d);
  // pull the gathered source row toward the WGP (L2-resident gather)
  __builtin_prefetch(h + (size_t)s * heads * C + lane * 4, 0, 1);
  for (int hd = 0; hd < heads; ++hd) {
    float alpha = w[(size_t)wave * heads + hd] / (den[d * heads + hd] + 1e-16f);
    const float* hp = h   + (size_t)s * heads * C + (size_t)hd * C;
    float*       ap = acc + (size_t)d * heads * C + (size_t)hd * C;
    for (int i = lane * 4; i < C; i += 128) {
      float4 hv = *(const float4*)(hp + i);
      atomicAdd(ap + i + 0, alpha * hv.x);
      atomicAdd(ap + i + 1, alpha * hv.y);
      atomicAdd(ap + i + 2, alpha * hv.z);
      atomicAdd(ap + i + 3, alpha * hv.w);
    }
  }
}

// ---------- bias + relu (in place) ----------
__global__ void bias_relu(float* __restrict__ acc, const float* __restrict__ b,
                          int total, int dim) {
  int t = blockIdx.x * blockDim.x + threadIdx.x;
  if (t >= total) return;
  float v = acc[t] + b[t % dim];
  acc[t] = v > 0.f ? v : 0.f;
}

// ---------- mean pool accumulate ----------
__global__ void pool_acc(const float* __restrict__ h, const int* __restrict__ batch,
                         float* __restrict__ pooled, float* __restrict__ cnt) {
  int t = blockIdx.x * blockDim.x + threadIdx.x;
  if (t >= N_NODES * HIDC) return;
  int node = t / HIDC, c = t % HIDC;
  int g = batch[node];
  atomicAdd(pooled + g * HIDC + c, h[t]);
  if (c == 0) atomicAdd(cnt + g, 1.0f);
}

// ---------- final FC (tiny: 64x64, K=128) ----------
__global__ void fc_kernel(const float* __restrict__ pooled, const float* __restrict__ cnt,
                          const float* __restrict__ w, const float* __restrict__ b,
                          float* __restrict__ out) {
  int t = blockIdx.x * blockDim.x + threadIdx.x;
  if (t >= NGR * OUT_C) return;
  int g = t / OUT_C, o = t % OUT_C;
  float inv = 1.0f / fmaxf(cnt[g], 1.0f);
  float acc = b[o];
  for (int k = 0; k < HIDC; ++k)
    acc += (pooled[g * HIDC + k] * inv) * w[k * OUT_C + o];
  out[t] = acc;
}

extern "C" void kernel_launch(void* const* d_in, const int* in_sizes, int n_in,
                              void* d_out, int out_size, void* d_ws, size_t ws_size,
                              hipStream_t stream) {
  const float* x    = (const float*)d_in[0];
  const int*   ei   = (const int*)  d_in[1];
  const int*   batch= (const int*)  d_in[2];
  const float* W1   = (const float*)d_in[3];
  const float* a1s  = (const float*)d_in[4];
  const float* a1d  = (const float*)d_in[5];
  const float* b1   = (const float*)d_in[6];
  const float* W2   = (const float*)d_in[7];
  const float* a2s  = (const float*)d_in[8];
  const float* a2d  = (const float*)d_in[9];
  const float* b2   = (const float*)d_in[10];
  const float* fcw  = (const float*)d_in[11];
  const float* fcb  = (const float*)d_in[12];
  float* out = (float*)d_out;

  // workspace carve-up (all offsets multiples of 16B)
  float* ws = (float*)d_ws;
  size_t off = 0;
  float*    h1   = ws + off; off += (size_t)N_NODES * H1DIM;   // GEMM1 output
  float*    as1  = ws + off; off += (size_t)N_NODES * HEADS;
  float*    ad1  = ws + off; off += (size_t)N_NODES * HEADS;
  unsigned* mx1  = (unsigned*)(ws + off); off += (size_t)N_NODES * HEADS;
  float*    den1 = ws + off; off += (size_t)N_NODES * HEADS;
  float*    w1e  = ws + off; off += (size_t)ETOT * HEADS;
  float*    acc1 = ws + off; off += (size_t)N_NODES * H1DIM;   // layer1 agg + relu (in place)
  float*    h2   = ws + off; off += (size_t)N_NODES * HIDC;    // GEMM2 output
  float*    as2  = ws + off; off += (size_t)N_NODES;
  float*    ad2  = ws + off; off += (size_t)N_NODES;
  unsigned* mx2  = (unsigned*)(ws + off); off += (size_t)N_NODES;
  float*    den2 = ws + off; off += (size_t)N_NODES;
  float*    w2e  = ws + off; off += (size_t)ETOT;
  float*    acc2 = ws + off; off += (size_t)N_NODES * HIDC;
  float*    pool = ws + off; off += (size_t)NGR * HIDC;
  float*    cnt  = ws + off; off += (size_t)NGR;

  const unsigned ORD_NEG_INF = 0x007FFFFFu;  // ordf(-inf)
  const int TB = 256;

  // (re)initialize accumulators every call
  fill_f32<<<CDIV(N_NODES * H1DIM, TB), TB, 0, stream>>>(acc1, 0.f, N_NODES * H1DIM);
  fill_f32<<<CDIV(N_NODES * HEADS, TB), TB, 0, stream>>>(den1, 0.f, N_NODES * HEADS);
  fill_u32<<<CDIV(N_NODES * HEADS, TB), TB, 0, stream>>>(mx1, ORD_NEG_INF, N_NODES * HEADS);
  fill_f32<<<CDIV(N_NODES * HIDC, TB), TB, 0, stream>>>(acc2, 0.f, N_NODES * HIDC);
  fill_f32<<<CDIV(N_NODES, TB), TB, 0, stream>>>(den2, 0.f, N_NODES);
  fill_u32<<<CDIV(N_NODES, TB), TB, 0, stream>>>(mx2, ORD_NEG_INF, N_NODES);
  fill_f32<<<CDIV(NGR * HIDC, TB), TB, 0, stream>>>(pool, 0.f, NGR * HIDC);
  fill_f32<<<CDIV(NGR, TB), TB, 0, stream>>>(cnt, 0.f, NGR);

  // ---- layer 1: GEMM (WMMA f32), logits, segment softmax, scatter, relu ----
  {
    int waves = (N_NODES / 16) * (H1DIM / 32);                  // 625 * 16
    gemm_wmma_f32<IN_C, H1DIM><<<CDIV(waves, 8), TB, 0, stream>>>(x, W1, h1, N_NODES);
  }
  alpha_kernel<<<CDIV(N_NODES * HEADS, 8), TB, 0, stream>>>(h1, a1s, a1d, as1, ad1,
                                                            N_NODES, HEADS, HIDC);
  edge_max<<<CDIV(ETOT * HEADS, TB), TB, 0, stream>>>(ei, as1, ad1, mx1, HEADS);
  edge_exp<<<CDIV(ETOT * HEADS, TB), TB, 0, stream>>>(ei, as1, ad1, mx1, w1e, den1, HEADS);
  edge_aggr<<<CDIV(ETOT, 8), TB, 0, stream>>>(ei, h1, w1e, den1, acc1, HEADS, HIDC);
  bias_relu<<<CDIV(N_NODES * H1DIM, TB), TB, 0, stream>>>(acc1, b1, N_NODES * H1DIM, H1DIM);

  // ---- layer 2 ----
  {
    int waves = (N_NODES / 16) * (HIDC / 32);                   // 625 * 4
    gemm_wmma_f32<H1DIM, HIDC><<<CDIV(waves, 8), TB, 0, stream>>>(acc1, W2, h2, N_NODES);
  }
  alpha_kernel<<<CDIV(N_NODES, 8), TB, 0, stream>>>(h2, a2s, a2d, as2, ad2,
                                                    N_NODES, 1, HIDC);
  edge_max<<<CDIV(ETOT, TB), TB, 0, stream>>>(ei, as2, ad2, mx2, 1);
  edge_exp<<<CDIV(ETOT, TB), TB, 0, stream>>>(ei, as2, ad2, mx2, w2e, den2, 1);
  edge_aggr<<<CDIV(ETOT, 8), TB, 0, stream>>>(ei, h2, w2e, den2, acc2, 1, HIDC);
  bias_relu<<<CDIV(N_NODES * HIDC, TB), TB, 0, stream>>>(acc2, b2, N_NODES * HIDC, HIDC);

  // ---- pool + FC ----
  pool_acc<<<CDIV(N_NODES * HIDC, TB), TB, 0, stream>>>(acc2, batch, pool, cnt);
  fc_kernel<<<CDIV(NGR * OUT_C, TB), TB, 0, stream>>>(pool, cnt, fcw, fcb, out);
}